// Pofusion_ONNX_25074019074128
// MI455X (gfx1250) — compile-verified
//
#include <hip/hip_runtime.h>
#include <hip/hip_bf16.h>

#define HH 768
#define WW 704
#define HW (HH * WW)          // 540672 per image
#define BB 2
#define NCLS 10
#define NSTUFF 6
#define NBOX 32
#define NTHING 4
#define RR 28
#define LL 39                 // 1 + NSTUFF + NBOX
#define TPB 256
#define BLOCKS_PER_IMG (HW / TPB)   // 2112, exact

// workspace layout (in int units)
#define WS_HIST 0             // BB * 512 ints (per image: 320 inst-vote + 6 stuff, padded)
#define WS_BOX  1024          // BB * 512 ints (32 boxes * 16 ints)
#define WS_LUT  2048          // BB * 40 ints

// ---- CDNA5 async global -> LDS (ASYNCcnt path), gfx1250 ----
__device__ __forceinline__ void async_g2l_b32(unsigned int lds_byte_off, const void* gptr) {
  unsigned long long ga = (unsigned long long)(size_t)gptr;
  asm volatile("global_load_async_to_lds_b32 %0, %1, off"
               :: "v"(lds_byte_off), "v"(ga) : "memory");
}
__device__ __forceinline__ void wait_asynccnt0() {
  asm volatile("s_wait_asynccnt 0x0" ::: "memory");
}

__device__ __forceinline__ int imin(int a, int b) { return a < b ? a : b; }
__device__ __forceinline__ int imax(int a, int b) { return a > b ? a : b; }

// ---------------- kernel 1: zero histograms + per-box params ----------------
__global__ void prep_kernel(const float* __restrict__ bbx, const int* __restrict__ cls,
                            int* __restrict__ ws) {
  int tid = threadIdx.x;
  for (int i = tid; i < 2 * 512; i += TPB) ws[WS_HIST + i] = 0;
  if (tid < BB * NBOX) {
    int b = tid >> 5, n = tid & 31;
    const float* bp = bbx + (size_t)(b * NBOX + n) * 4;
    float by0 = bp[0], bx0 = bp[1], by1 = bp[2], bx1 = bp[3];
    int y0 = (int)floorf(by0), x0 = (int)floorf(bx0);
    int y1 = (int)floorf(by1), x1 = (int)floorf(bx1);
    float h = fmaxf((float)(y1 - y0 + 1), 1.0f);
    float w = fmaxf((float)(x1 - x0 + 1), 1.0f);
    int iy1 = (int)(rintf(by1) + 1.0f);   // ties-to-even, matches jnp.round
    int ix1 = (int)(rintf(bx1) + 1.0f);
    int* o = ws + WS_BOX + (b * NBOX + n) * 16;
    o[0] = y0; o[1] = x0; o[2] = y1; o[3] = x1;
    o[4] = iy1; o[5] = ix1; o[6] = cls[b * NBOX + n]; o[7] = 0;
    o[8] = __float_as_int((float)RR / h);
    o[9] = __float_as_int((float)RR / w);
    o[10] = 0; o[11] = 0; o[12] = 0; o[13] = 0; o[14] = 0; o[15] = 0;
  }
}

// ------------- kernel 2: fused logits, argmaxes, histograms, pp -------------
__global__ void __launch_bounds__(TPB) fuse_kernel(const float* __restrict__ sem,
                                                   const float* __restrict__ roi,
                                                   int* __restrict__ ws,
                                                   int* __restrict__ out) {
  __shared__ float s_sem[NCLS * TPB];   // 10 KB: 10 channels x 256 pixels
  __shared__ int   s_box[NBOX * 16];    // 2 KB
  __shared__ int   s_hist[326];         // 320 inst votes + 6 stuff

  const int tid = threadIdx.x;
  const int b   = blockIdx.x / BLOCKS_PER_IMG;
  const int pix = (blockIdx.x - b * BLOCKS_PER_IMG) * TPB + tid;

  // Async-stage the 10 semantic channels for this pixel tile into LDS.
  const float* semb = sem + (size_t)b * NCLS * HW + pix;
  unsigned int lbase = (unsigned int)(size_t)(&s_sem[tid]);   // low 32 bits = LDS byte addr
#pragma unroll
  for (int c = 0; c < NCLS; ++c)
    async_g2l_b32(lbase + (unsigned)c * (TPB * 4), semb + (size_t)c * HW);

  for (int i = tid; i < NBOX * 16; i += TPB) s_box[i] = ws[WS_BOX + b * 512 + i];
  for (int i = tid; i < 326; i += TPB) s_hist[i] = 0;

  wait_asynccnt0();
  __syncthreads();

  // argmax over all 10 sem channels (first-max tie-break)
  float bs = s_sem[tid]; int sp = 0;
#pragma unroll
  for (int c = 1; c < NCLS; ++c) { float v = s_sem[c * TPB + tid]; if (v > bs) { bs = v; sp = c; } }

  // fused argmax: channels 0..5 = stuff sem logits
  float bf = s_sem[tid]; int pp = 0;
#pragma unroll
  for (int c = 1; c < NSTUFF; ++c) { float v = s_sem[c * TPB + tid]; if (v > bf) { bf = v; pp = c; } }

  const int y = pix / WW;
  const int x = pix - y * WW;

  for (int n = 0; n < NBOX; ++n) {
    const int* bi = &s_box[n * 16];
    const int y0 = bi[0], x0 = bi[1], y1 = bi[2], x1 = bi[3];
    const int iy1 = bi[4], ix1 = bi[5], cl = bi[6];
    const bool inP = (y >= imax(y0, 0)) & (y < imin(y1 + 1, HH)) &
                     (x >= imax(x0, 0)) & (x < imin(x1 + 1, WW));
    const bool inI = (y >= y0) & (y < iy1) & (x >= x0) & (x < ix1);
    float vc = -0.0f;   // (sig(-100)+sig(-100))*(-200) underflows to -0
    if (inP | inI) {
      float pm = -100.0f, pi = -100.0f;
      if (inI) pi = s_sem[(NSTUFF + cl) * TPB + tid];   // dynamic channel: cheap LDS gather
      if (inP) {
        const float rh = __int_as_float(bi[8]);
        const float rw = __int_as_float(bi[9]);
        float sy = fminf(fmaxf(((float)(y - y0) + 0.5f) * rh - 0.5f, 0.0f), 27.0f);
        float sx = fminf(fmaxf(((float)(x - x0) + 0.5f) * rw - 0.5f, 0.0f), 27.0f);
        int ylo = (int)sy, xlo = (int)sx;
        int yhi = imin(ylo + 1, RR - 1), xhi = imin(xlo + 1, RR - 1);
        float wy = sy - (float)ylo, wx = sx - (float)xlo;
        const float* m = roi + ((size_t)((b * NBOX + n) * NTHING + cl)) * (RR * RR);
        float a00 = m[ylo * RR + xlo], a01 = m[ylo * RR + xhi];
        float a10 = m[yhi * RR + xlo], a11 = m[yhi * RR + xhi];
        float r0 = a00 + (a01 - a00) * wx;
        float r1 = a10 + (a11 - a10) * wx;
        pm = r0 + (r1 - r0) * wy;
      }
      float sgi = 1.0f / (1.0f + __expf(-pi));
      float sgm = 1.0f / (1.0f + __expf(-pm));
      vc = (sgi + sgm) * (pi + pm);
    }
    if (vc > bf) { bf = vc; pp = NSTUFF + n; }   // strict > keeps first-max semantics
  }

  // histograms (LDS-privatized)
  if (pp >= NSTUFF) atomicAdd(&s_hist[(pp - NSTUFF) * NCLS + sp], 1);
  else              atomicAdd(&s_hist[320 + pp], 1);

  out[(size_t)b * HW + pix] = pp;   // stash pp in the seam output slot; pass 2 rewrites it

  __syncthreads();
  for (int i = tid; i < 326; i += TPB) {
    int v = s_hist[i];
    if (v) atomicAdd(&ws[WS_HIST + b * 512 + i], v);
  }
}

// ---------------- kernel 3: per-image labeling -> LUT + po_cls ---------------
__global__ void label_kernel(const int* __restrict__ cls_in, int* __restrict__ ws,
                             int* __restrict__ out) {
  if (threadIdx.x != 0) return;
  const int b = blockIdx.x;
  const int* hist = ws + WS_HIST + b * 512;

  int rowsum[NBOX], maj[NBOX], cmax[NBOX];
  for (int n = 0; n < NBOX; ++n) {
    int m = hist[n * NCLS], mi = 0, s = m;
    for (int c = 1; c < NCLS; ++c) {
      int v = hist[n * NCLS + c]; s += v;
      if (v > m) { m = v; mi = c; }
    }
    rowsum[n] = s; maj[n] = mi; cmax[n] = m;
  }

  int idx[NBOX]; int run = 0;
  for (int n = 0; n < NBOX; ++n) { if (rowsum[n] > 0) run++; idx[n] = run - 1; }

  int sem_val[NBOX], inst_val[NBOX]; bool to_stuff[NBOX];
  for (int n = 0; n < NBOX; ++n) {
    float frac = (float)cmax[n] / (float)imax(rowsum[n], 1);
    int thing = cls_in[b * NBOX + n] + NSTUFF;
    bool match = (maj[n] == thing);
    to_stuff[n] = (!match) && (frac >= 0.5f) && (maj[n] < NSTUFF);
    sem_val[n]  = to_stuff[n] ? maj[n] : thing;
    inst_val[n] = to_stuff[n] ? -1 : idx[n];
  }

  int sh[NSTUFF];
  for (int s = 0; s < NSTUFF; ++s) sh[s] = hist[320 + s];
  for (int n = 0; n < NBOX; ++n) if (to_stuff[n]) sh[maj[n]] += rowsum[n];

  int s_rank[NSTUFF]; run = 0;
  for (int s = 0; s < NSTUFF; ++s) { if (sh[s] > 0) run++; s_rank[s] = run - 1; }
  const int n_stuff = run;

  int ih[NBOX];
  for (int i = 0; i < NBOX; ++i) ih[i] = 0;
  for (int n = 0; n < NBOX; ++n)
    if (!to_stuff[n] && rowsum[n] > 0) ih[idx[n]] += rowsum[n];

  int i_rank[NBOX]; run = 0;
  for (int i = 0; i < NBOX; ++i) { if (ih[i] > 0) run++; i_rank[i] = run - 1; }

  // seam LUT: seam is a pure per-image function of pp
  int* lut = ws + WS_LUT + b * 40;
  for (int s = 0; s < NSTUFF; ++s) lut[s] = 1 + s_rank[s];
  for (int n = 0; n < NBOX; ++n) {
    int v;
    if (inst_val[n] >= 0)            v = 1 + n_stuff + i_rank[inst_val[n]];
    else if (sem_val[n] < NSTUFF)    v = 1 + s_rank[sem_val[n]];
    else                             v = 0;
    lut[NSTUFF + n] = v;
  }

  // po_cls
  int co[LL];
  for (int k = 0; k < LL; ++k) co[k] = 255;
  for (int s = 0; s < NSTUFF; ++s) if (sh[s] > 0) co[1 + s_rank[s]] = s;
  int inst_sem[NBOX];
  for (int i = 0; i < NBOX; ++i) inst_sem[i] = 255;
  for (int n = 0; n < NBOX; ++n)
    if (rowsum[n] > 0 && !to_stuff[n]) inst_sem[idx[n]] = sem_val[n];
  for (int i = 0; i < NBOX; ++i)
    if (ih[i] > 0) co[1 + n_stuff + i_rank[i]] = inst_sem[i];

  int* pcls = out + (size_t)BB * HW + b * LL;
  int* pic  = out + (size_t)BB * HW + BB * LL + b * LL;
  for (int k = 0; k < LL; ++k) { pcls[k] = co[k]; pic[k] = 0; }
}

// ---------------- kernel 4: LUT rewrite pp -> seam (in place) ----------------
__global__ void __launch_bounds__(TPB) seam_kernel(const int* __restrict__ ws,
                                                   int* __restrict__ out) {
  __shared__ int lut[40];
  const int tid = threadIdx.x;
  const int b = blockIdx.x / BLOCKS_PER_IMG;
  if (tid < NSTUFF + NBOX) lut[tid] = ws[WS_LUT + b * 40 + tid];
  __syncthreads();
  size_t i = (size_t)b * HW + (size_t)(blockIdx.x - b * BLOCKS_PER_IMG) * TPB + tid;
  out[i] = lut[out[i]];
}

extern "C" void kernel_launch(void* const* d_in, const int* in_sizes, int n_in,
                              void* d_out, int out_size, void* d_ws, size_t ws_size,
                              hipStream_t stream) {
  const float* sem = (const float*)d_in[0];   // (2,10,768,704) f32
  const float* roi = (const float*)d_in[1];   // (2,32,4,28,28) f32
  const float* bbx = (const float*)d_in[2];   // (2,32,4) f32
  const int*   cls = (const int*)d_in[3];     // (2,32) i32
  // d_in[4] = img_size (constant {768,704}); baked into kernels.
  int* ws  = (int*)d_ws;
  int* out = (int*)d_out;   // int32 outputs written through int*

  prep_kernel <<<1, TPB, 0, stream>>>(bbx, cls, ws);
  fuse_kernel <<<BB * BLOCKS_PER_IMG, TPB, 0, stream>>>(sem, roi, ws, out);
  label_kernel<<<BB, 32, 0, stream>>>(cls, ws, out);
  seam_kernel <<<BB * BLOCKS_PER_IMG, TPB, 0, stream>>>(ws, out);
}